// GCNMLP_81320910782821
// MI455X (gfx1250) — compile-verified
//
#include <hip/hip_runtime.h>
#include <hip/hip_bf16.h>

// ---------------- problem constants ----------------
#define BATCH   32
#define TSTEPS  12
#define ENODE   10000
#define DIN     4
#define CDIM    128
#define HDIM    64
#define NPRED   12
#define PD      4
#define NEDGE   160000
#define NROWS   (BATCH * ENODE)      // 320000, multiple of 16

typedef __attribute__((ext_vector_type(2))) float v2f;
typedef __attribute__((ext_vector_type(8))) float v8f;

// V_WMMA_F32_16X16X4_F32 : D(16x16 f32) = A(16x4 f32) * B(4x16 f32) + C
__device__ __forceinline__ v8f wmma4(v2f a, v2f b, v8f c) {
    return __builtin_amdgcn_wmma_f32_16x16x4_f32(
        /*neg_a=*/false, a, /*neg_b=*/false, b,
        /*c_mod=*/(short)0, c, /*reuse_a=*/false, /*reuse_b=*/false);
}

// ---------------- utility kernels ----------------
__global__ void k_zero4(float4* __restrict__ p, long n4) {
    long i = (long)blockIdx.x * blockDim.x + threadIdx.x;
    if (i < n4) p[i] = make_float4(0.f, 0.f, 0.f, 0.f);
}

__global__ void k_deg(const int* __restrict__ row, int* __restrict__ cnt, int n) {
    int i = blockIdx.x * blockDim.x + threadIdx.x;
    if (i < n) atomicAdd(&cnt[row[i]], 1);
}

__global__ void k_dis(const int* __restrict__ cnt, float* __restrict__ dis, int n) {
    int i = blockIdx.x * blockDim.x + threadIdx.x;
    if (i < n) {
        int c = cnt[i];
        dis[i] = (c > 0) ? rsqrtf((float)c) : 0.f;
    }
}

__global__ void k_wgt(const int* __restrict__ row, const int* __restrict__ col,
                      const float* __restrict__ dis, float* __restrict__ wgt, int n) {
    int i = blockIdx.x * blockDim.x + threadIdx.x;
    if (i < n) wgt[i] = -dis[row[i]] * dis[col[i]];
}

// ---------------- temporal conv + embedding ----------------
// one wave per (b,e): x3[o] = sum_{t,i} x[b,t,e,i]*conv_w[o,i,t]; h = x3@embed_w + eb
__global__ __launch_bounds__(256) void k_embed(
    const float* __restrict__ x, const float* __restrict__ cw,
    const float* __restrict__ cb, const float* __restrict__ ew,
    const float* __restrict__ eb, float* __restrict__ h) {
    __shared__ float xs[8][52];              // 48 inputs + 4 conv outputs per wave
    const int wave = threadIdx.x >> 5;
    const int lane = threadIdx.x & 31;
    const long pe = (long)blockIdx.x * 8 + wave;   // (b,e) flat, grid sized exactly
    const int b = (int)(pe / ENODE);
    const int e = (int)(pe % ENODE);
    for (int j = lane; j < 48; j += 32) {
        int t = j >> 2, i = j & 3;
        xs[wave][j] = x[(((size_t)b * TSTEPS + t) * ENODE + e) * DIN + i];
    }
    __syncthreads();
    if (lane < DIN) {
        float s = 0.f;
        #pragma unroll
        for (int t = 0; t < TSTEPS; ++t)
            #pragma unroll
            for (int i = 0; i < DIN; ++i)
                s += xs[wave][t * 4 + i] * cw[(lane * DIN + i) * TSTEPS + t];
        xs[wave][48 + lane] = s + cb[lane];
    }
    __syncthreads();
    const float x0 = xs[wave][48], x1 = xs[wave][49];
    const float x2 = xs[wave][50], x3 = xs[wave][51];
    #pragma unroll
    for (int c = lane; c < CDIM; c += 32) {
        float v = eb[c] + x0 * ew[c] + x1 * ew[CDIM + c]
                        + x2 * ew[2 * CDIM + c] + x3 * ew[3 * CDIM + c];
        h[pe * CDIM + c] = v;
    }
}

// ---------------- edge aggregation (scatter-add, L2 resident) ----------------
// thread -> (edge, b, c4): tx1[b,row,:] += w * h[b,col,:]
__global__ __launch_bounds__(256) void k_agg(
    const float* __restrict__ h, const int* __restrict__ row,
    const int* __restrict__ col, const float* __restrict__ wgt,
    float* __restrict__ tx1) {
    long t = (long)blockIdx.x * 256 + threadIdx.x;
    int c4 = (int)(t & 31);
    int b  = (int)((t >> 5) & 31);
    long e = t >> 10;
    if (e >= NEDGE) return;
    float w = wgt[e];
    if (w == 0.f) return;
    int r = row[e], c = col[e];
    const float4 v = *(const float4*)(h + ((size_t)b * ENODE + c) * CDIM + c4 * 4);
    float* dst = tx1 + ((size_t)b * ENODE + r) * CDIM + c4 * 4;
    atomicAdd(dst + 0, w * v.x);
    atomicAdd(dst + 1, w * v.y);
    atomicAdd(dst + 2, w * v.z);
    atomicAdd(dst + 3, w * v.w);
}

// ---------------- Cheb layer GEMM: hout = relu(h@W0 + tx1@W1 + b), in place ----
// block = 16 rows, 8 waves, wave w -> cols [16w,16w+16)
__global__ __launch_bounds__(256, 2) void k_cheb(
    const float* __restrict__ h, const float* __restrict__ tx1,
    const float* __restrict__ W0, const float* __restrict__ W1,
    const float* __restrict__ bias, float* __restrict__ hout) {
    __shared__ float sH[16 * CDIM];
    __shared__ float sT[16 * CDIM];
    const int row0 = blockIdx.x * 16;
    const int tid = threadIdx.x;
    {   // coalesced float4 staging of 16x128 A-tiles
        const float4* gH = (const float4*)(h   + (size_t)row0 * CDIM);
        const float4* gT = (const float4*)(tx1 + (size_t)row0 * CDIM);
        float4* s4H = (float4*)sH;
        float4* s4T = (float4*)sT;
        for (int i = tid; i < 16 * CDIM / 4; i += 256) { s4H[i] = gH[i]; s4T[i] = gT[i]; }
    }
    __syncthreads();
    const int wave = tid >> 5;
    const int lane = tid & 31;
    const int m  = lane & 15;
    const int hi = lane >> 4;
    const int col0 = wave * 16;
    const int n = m;
    v8f acc = {};
    #pragma unroll 4
    for (int kk = 0; kk < CDIM; kk += 4) {
        const int k0 = kk + 2 * hi;
        v2f aH; aH[0] = sH[m * CDIM + k0]; aH[1] = sH[m * CDIM + k0 + 1];
        v2f b0; b0[0] = W0[(size_t)k0 * CDIM + col0 + n];
                b0[1] = W0[(size_t)(k0 + 1) * CDIM + col0 + n];
        acc = wmma4(aH, b0, acc);
        v2f aT; aT[0] = sT[m * CDIM + k0]; aT[1] = sT[m * CDIM + k0 + 1];
        v2f b1; b1[0] = W1[(size_t)k0 * CDIM + col0 + n];
                b1[1] = W1[(size_t)(k0 + 1) * CDIM + col0 + n];
        acc = wmma4(aT, b1, acc);
    }
    const float bc = bias[col0 + n];
    #pragma unroll
    for (int r = 0; r < 8; ++r) {
        const int mm = r + 8 * hi;
        float v = acc[r] + bc;
        v = v > 0.f ? v : 0.f;
        hout[((size_t)(row0 + mm)) * CDIM + col0 + n] = v;
    }
}

// ---------------- fused MLP: z=relu(h@W1+b1); preds=z@W2+b2; transpose-store ----
// block = 16 rows, 4 waves. Phase1: 4x 16-col tiles of z (64) into LDS.
// Phase2: 3 waves produce 48 output cols, scattered to [B,12,E,4].
__global__ __launch_bounds__(128, 2) void k_mlp(
    const float* __restrict__ h, const float* __restrict__ W1,
    const float* __restrict__ b1, const float* __restrict__ W2,
    const float* __restrict__ b2, float* __restrict__ out) {
    __shared__ float sA[16 * CDIM];
    __shared__ float sZ[16 * HDIM];
    const int row0 = blockIdx.x * 16;
    const int tid = threadIdx.x;
    {
        const float4* g = (const float4*)(h + (size_t)row0 * CDIM);
        float4* s4 = (float4*)sA;
        for (int i = tid; i < 16 * CDIM / 4; i += 128) s4[i] = g[i];
    }
    __syncthreads();
    const int wave = tid >> 5;
    const int lane = tid & 31;
    const int m  = lane & 15;
    const int hi = lane >> 4;
    const int n = m;
    {   // phase 1: hidden layer, all 4 waves
        const int col0 = wave * 16;
        v8f acc = {};
        #pragma unroll 4
        for (int kk = 0; kk < CDIM; kk += 4) {
            const int k0 = kk + 2 * hi;
            v2f a; a[0] = sA[m * CDIM + k0]; a[1] = sA[m * CDIM + k0 + 1];
            v2f b; b[0] = W1[(size_t)k0 * HDIM + col0 + n];
                   b[1] = W1[(size_t)(k0 + 1) * HDIM + col0 + n];
            acc = wmma4(a, b, acc);
        }
        const float bc = b1[col0 + n];
        #pragma unroll
        for (int r = 0; r < 8; ++r) {
            const int mm = r + 8 * hi;
            float v = acc[r] + bc;
            v = v > 0.f ? v : 0.f;
            sZ[mm * HDIM + col0 + n] = v;
        }
    }
    __syncthreads();
    if (wave < 3) {   // phase 2: output layer (48 cols), wave-uniform branch
        const int col0 = wave * 16;
        v8f acc = {};
        #pragma unroll 4
        for (int kk = 0; kk < HDIM; kk += 4) {
            const int k0 = kk + 2 * hi;
            v2f a; a[0] = sZ[m * HDIM + k0]; a[1] = sZ[m * HDIM + k0 + 1];
            v2f b; b[0] = W2[(size_t)k0 * (NPRED * PD) + col0 + n];
                   b[1] = W2[(size_t)(k0 + 1) * (NPRED * PD) + col0 + n];
            acc = wmma4(a, b, acc);
        }
        const float bc = b2[col0 + n];
        #pragma unroll
        for (int r = 0; r < 8; ++r) {
            const int mm = r + 8 * hi;
            const int R = row0 + mm;
            const int bb = R / ENODE;
            const int e  = R % ENODE;
            const int j = col0 + n;
            const int p = j >> 2, d = j & 3;
            out[(((size_t)bb * NPRED + p) * ENODE + e) * PD + d] = acc[r] + bc;
        }
    }
}

// ---------------- host-side orchestration ----------------
extern "C" void kernel_launch(void* const* d_in, const int* in_sizes, int n_in,
                              void* d_out, int out_size, void* d_ws, size_t ws_size,
                              hipStream_t stream) {
    (void)in_sizes; (void)n_in; (void)out_size; (void)ws_size;
    const float* x        = (const float*)d_in[0];
    const int*   eidx     = (const int*)d_in[1];       // [2, NEDGE]
    const float* conv_w   = (const float*)d_in[2];
    const float* conv_b   = (const float*)d_in[3];
    const float* embed_w  = (const float*)d_in[4];
    const float* embed_b  = (const float*)d_in[5];
    const float* c0_w0    = (const float*)d_in[6];
    const float* c0_w1    = (const float*)d_in[7];
    const float* c0_b     = (const float*)d_in[8];
    const float* c1_w0    = (const float*)d_in[9];
    const float* c1_w1    = (const float*)d_in[10];
    const float* c1_b     = (const float*)d_in[11];
    const float* mlp_w1   = (const float*)d_in[12];
    const float* mlp_b1   = (const float*)d_in[13];
    const float* mlp_w2   = (const float*)d_in[14];
    const float* mlp_b2   = (const float*)d_in[15];
    float* out = (float*)d_out;

    const int* row = eidx;
    const int* col = eidx + NEDGE;

    // workspace layout (bytes)
    char* ws = (char*)d_ws;
    const size_t HBYTES = (size_t)NROWS * CDIM * sizeof(float);   // 163,840,000
    float* h    = (float*)(ws);
    float* tx1  = (float*)(ws + HBYTES);
    int*   cnt  = (int*)  (ws + 2 * HBYTES);
    float* dis  = (float*)(ws + 2 * HBYTES + 16384 * sizeof(int));
    float* wgt  = (float*)(ws + 2 * HBYTES + 16384 * sizeof(int) + 16384 * sizeof(float));

    const long tx1_f4 = (long)NROWS * CDIM / 4;      // 10,240,000 float4

    // 1) degree -> dis -> edge weights
    k_zero4<<<(16384 / 4 + 255) / 256, 256, 0, stream>>>((float4*)cnt, 16384 / 4);
    k_deg  <<<(NEDGE + 255) / 256, 256, 0, stream>>>(row, cnt, NEDGE);
    k_dis  <<<(ENODE + 255) / 256, 256, 0, stream>>>(cnt, dis, ENODE);
    k_wgt  <<<(NEDGE + 255) / 256, 256, 0, stream>>>(row, col, dis, wgt, NEDGE);

    // 2) temporal conv + embedding -> h [B,E,C]
    k_embed<<<NROWS / 8, 256, 0, stream>>>(x, conv_w, conv_b, embed_w, embed_b, h);

    // 3) Cheb layer 0
    k_zero4<<<(int)((tx1_f4 + 255) / 256), 256, 0, stream>>>((float4*)tx1, tx1_f4);
    k_agg  <<<(int)(((long)NEDGE * 1024) / 256), 256, 0, stream>>>(h, row, col, wgt, tx1);
    k_cheb <<<NROWS / 16, 256, 0, stream>>>(h, tx1, c0_w0, c0_w1, c0_b, h);

    // 4) Cheb layer 1
    k_zero4<<<(int)((tx1_f4 + 255) / 256), 256, 0, stream>>>((float4*)tx1, tx1_f4);
    k_agg  <<<(int)(((long)NEDGE * 1024) / 256), 256, 0, stream>>>(h, row, col, wgt, tx1);
    k_cheb <<<NROWS / 16, 256, 0, stream>>>(h, tx1, c1_w0, c1_w1, c1_b, h);

    // 5) fused MLP + output transpose
    k_mlp  <<<NROWS / 16, 128, 0, stream>>>(h, mlp_w1, mlp_b1, mlp_w2, mlp_b2, out);
}